// Sparse3TEncoder_49873160241491
// MI455X (gfx1250) — compile-verified
//
#include <hip/hip_runtime.h>
#include <cstdint>
#include <cstddef>

// Sparse 3D+T multi-resolution encoder (gather-bound).
// 8 lanes per query (lane = feature f), 32 queries per 256-thread block.
// Branchless parity-selected corners; async global->LDS staging of positions;
// LDS transpose + nontemporal b128 stores for the streaming output.

#define N_QUERIES 500000
#define QPB 32          // queries per block
#define THREADS 256     // 8 lanes per query

typedef float v4f __attribute__((ext_vector_type(4)));

__global__ __launch_bounds__(THREADS) void sparse3t_encode(
    const float4* __restrict__ pos,
    const float*  __restrict__ feats,
    const int* __restrict__ s0,
    const int* __restrict__ s1,
    const int* __restrict__ s2,
    const int* __restrict__ s3,
    float* __restrict__ out)
{
    __shared__ float4 sPos[QPB];        // 512 B
    __shared__ float  sOut[QPB * 32];   // 4 KB

    const int tid    = threadIdx.x;
    const int blockQ = blockIdx.x * QPB;

    // ---- Stage this block's 32 positions into LDS via the CDNA5 async DMA
    // path (ASYNCcnt-tracked, bypasses VGPRs). Wave 0 issues one
    // global_load_async_to_lds_b128 covering 32 lanes x 16B = 512B.
    if (tid < QPB) {
        const float4* gsrc   = pos + blockQ + tid;
        unsigned      ldsoff = (unsigned)(size_t)(&sPos[tid]); // low 32b of generic ptr = LDS offset
        asm volatile("global_load_async_to_lds_b128 %0, %1, off"
                     :
                     : "v"(ldsoff), "v"(gsrc)
                     : "memory");
        asm volatile("s_wait_asynccnt 0" ::: "memory");
    }
    __syncthreads();

    const int ql = tid >> 3;   // local query 0..31
    const int f  = tid & 7;    // feature index 0..7 (16B slice of the 128B row)

    const float4 P = sPos[ql];

    // Cubic Lagrange temporal basis at nodes {0, 1/3, 2/3, 1}.
    const float t  = P.w;
    const float a0 = t;
    const float a1 = t - (1.0f / 3.0f);
    const float a2 = t - (2.0f / 3.0f);
    const float a3 = t - 1.0f;
    const float V0 = a1 * a2 * a3 * -4.5f;
    const float V1 = a0 * a2 * a3 *  13.5f;
    const float V2 = a0 * a1 * a3 * -13.5f;
    const float V3 = a0 * a1 * a2 *   4.5f;

    const int* serials[4] = {s0, s1, s2, s3};
    const int  resA[4]    = {17, 33, 65, 129};

    float acc[4];

    #pragma unroll
    for (int L = 0; L < 4; ++L) {
        const int   R      = resA[L];
        const float inv_dx = (float)(R - 1);
        const float px = P.x * inv_dx;
        const float py = P.y * inv_dx;
        const float pz = P.z * inv_dx;
        const int bx = (int)floorf(px);
        const int by = (int)floorf(py);
        const int bz = (int)floorf(pz);
        const float fx = px - (float)bx;
        const float fy = py - (float)by;
        const float fz = pz - (float)bz;
        // Only corners with (i+j+k) even carry a serial: exactly the 4 corners
        // with u = (par + v + w) & 1, (v,w) in {0,1}^2, par = (bx+by+bz)&1.
        const int par = (bx + by + bz) & 1;
        const int* __restrict__ ser = serials[L];

        int   row[4];
        float wgt[4];
        #pragma unroll
        for (int c = 0; c < 4; ++c) {
            const int v = c & 1;
            const int w = (c >> 1) & 1;
            const int u = (par + v + w) & 1;
            const int ix = bx + u, iy = by + v, iz = bz + w;
            const bool inb = ((unsigned)ix < (unsigned)R) &
                             ((unsigned)iy < (unsigned)R) &
                             ((unsigned)iz < (unsigned)R);
            // Clamp (matches reference jnp.clip) and load unconditionally.
            const int icx = min(max(ix, 0), R - 1);
            const int icy = min(max(iy, 0), R - 1);
            const int icz = min(max(iz, 0), R - 1);
            const int s   = ser[(icx * R + icy) * R + icz];
            const bool valid = inb & (s > 0);
            const float wu = u ? fx : 1.0f - fx;
            const float wv = v ? fy : 1.0f - fy;
            const float ww = w ? fz : 1.0f - fz;
            wgt[c] = valid ? wu * wv * ww : 0.0f;
            row[c] = max(s - 1, 0);            // matches reference jnp.maximum
        }

        float aL = 0.0f;
        #pragma unroll
        for (int c = 0; c < 4; ++c) {
            // 8 lanes of this query together fetch the contiguous 128B row.
            const v4f fv =
                *(const v4f*)(feats + (size_t)row[c] * 32 + f * 4);
            const float tf = fv.x * V0 + fv.y * V1 + fv.z * V2 + fv.w * V3;
            aL = fmaf(wgt[c], tf, aL);
        }
        acc[L] = aL;
    }

    // ---- Transpose through LDS so the global store is pure b128, contiguous.
    #pragma unroll
    for (int L = 0; L < 4; ++L)
        sOut[ql * 32 + L * 8 + f] = acc[L];

    __syncthreads();

    // 32 queries * 32 floats = 256 float4 = one per thread, fully coalesced.
    // Streaming output: nontemporal store keeps L2 for the feats/serial tables.
    v4f*       outv  = (v4f*)out;
    const v4f* sOutV = (const v4f*)sOut;
    __builtin_nontemporal_store(sOutV[tid], &outv[(size_t)blockQ * 8 + tid]);
}

extern "C" void kernel_launch(void* const* d_in, const int* in_sizes, int n_in,
                              void* d_out, int out_size, void* d_ws, size_t ws_size,
                              hipStream_t stream) {
    const float4* pos   = (const float4*)d_in[0];
    const float*  feats = (const float*)d_in[1];
    const int*    s0    = (const int*)d_in[2];
    const int*    s1    = (const int*)d_in[3];
    const int*    s2    = (const int*)d_in[4];
    const int*    s3    = (const int*)d_in[5];
    float*        out   = (float*)d_out;

    const int nblocks = N_QUERIES / QPB; // 500000 / 32 = 15625, exact
    sparse3t_encode<<<nblocks, THREADS, 0, stream>>>(pos, feats, s0, s1, s2, s3, out);
}